// DistributeAttention_50199577755806
// MI455X (gfx1250) — compile-verified
//
#include <hip/hip_runtime.h>
#include <stdint.h>

// PSANet 'distribute': out[n, h*W+w, i, j] = x[n, (i-h+63)*127 + (j-w+63), h, w]
// N=2, H=W=64, HM=WM=127. Pure gather, zero FLOPs, bandwidth-bound:
// min traffic = 134 MB read + 134 MB write = 268 MB -> ~11.5 us @ 23.3 TB/s.
//
// One workgroup per (n,h,i); a = i-h+63 fixed. Stage exactly the needed
// trapezoid {x[n, a*127+b, h, w] : j=b+w-63 in [0,64)} into LDS using CDNA5
// per-lane async global->LDS loads (coalesced contiguous row segments,
// tracked by ASYNCcnt), then conflict-free transposed LDS reads and
// coalesced non-temporal global stores. Loading the full 127x64 rectangle
// (e.g. one TDM tile) would cost 2x read traffic; the predicated per-lane
// async gather moves exactly 4096 of the 8128 tile elements.

namespace {
constexpr int Hf  = 64;
constexpr int Wf  = 64;
constexpr int HMm = 2 * Hf - 1;   // 127
constexpr int WMm = 2 * Wf - 1;   // 127
constexpr int TP  = Wf + 1;       // 65: padded LDS row stride -> conflict-free
}

__global__ __launch_bounds__(256)
void psa_distribute_kernel(const float* __restrict__ x, float* __restrict__ out)
{
    __shared__ float tile[HMm * TP];          // 127 * 65 * 4 = 33,020 B LDS

    const int bid = blockIdx.x;               // = (n*64 + h)*64 + i
    const int n   = bid >> 12;
    const int h   = (bid >> 6) & 63;
    const int i   = bid & 63;
    const int a   = i - h + (Hf - 1);         // 0..126

    const int t      = threadIdx.x;           // 256 threads = 8 wave32
    const int lane64 = t & 63;                // 'w' in phase 1, 'j' in phase 2
    const int quad   = t >> 6;                // 0..3

    // &x[n][a*127 + 0][h][0]; row b lives at src + b*4096, contiguous in w.
    const float* src = x +
        ((((size_t)n * (HMm * WMm) + (size_t)a * WMm) * Hf + h) * Wf);

    // ---------------- Phase 1: async gather of exactly the needed trapezoid.
    // Row b needs w with j = b + w - 63 in [0,64): one contiguous segment,
    // so each wave's active lanes form a coalesced burst. EXEC predication
    // is honored per-lane by GLOBAL_LOAD_ASYNC_TO_LDS_B32.
    {
        const int  w  = lane64;
        const float* gp0 = src + (size_t)quad * (Hf * Wf) + w;
        uint32_t lds0 = (uint32_t)(size_t)(const void*)&tile[quad * TP + w];
#pragma unroll
        for (int k = 0; k < 32; ++k) {
            const int b = quad + 4 * k;       // quad, quad+4, ..., quad+124
            if (b < HMm && (unsigned)(b + w - (Hf - 1)) < (unsigned)Wf) {
                const float* gp = gp0 + (size_t)(4 * k) * (Hf * Wf);
                uint32_t lds_off = lds0 + (uint32_t)(4 * k) * (TP * 4);
                asm volatile("global_load_async_to_lds_b32 %0, %1, off"
                             :: "v"(lds_off), "v"(gp)
                             : "memory");
            }
        }
    }

#if __has_builtin(__builtin_amdgcn_s_wait_asynccnt)
    __builtin_amdgcn_s_wait_asynccnt(0);
#else
    asm volatile("s_wait_asynccnt 0" ::: "memory");
#endif
    __syncthreads();

    // ---------------- Phase 2: transposed LDS reads, coalesced NT stores.
    // out flat = (n*4096 + h*64)*4096 + w*4096 + i*64 + j
    const int j = lane64;
    float* dst = out + (size_t)(n * 4096 + h * 64) * 4096 + (size_t)(i * 64);

#pragma unroll
    for (int k = 0; k < 16; ++k) {
        const int w = quad + 4 * k;
        const int b = j - w + (Wf - 1);       // 0..126
        // LDS addr = b*65 + w: consecutive-j lanes stride 65 floats ->
        // 64 distinct banks -> conflict-free.
        float v = tile[b * TP + w];
        __builtin_nontemporal_store(v, dst + (size_t)w * (Hf * Wf) + j);
    }
}

extern "C" void kernel_launch(void* const* d_in, const int* in_sizes, int n_in,
                              void* d_out, int out_size, void* d_ws, size_t ws_size,
                              hipStream_t stream)
{
    (void)in_sizes; (void)n_in; (void)out_size; (void)d_ws; (void)ws_size;
    const float* x = (const float*)d_in[0];
    float* out     = (float*)d_out;

    // One block per (n, h, i): 2 * 64 * 64 = 8192 blocks, 256 threads.
    psa_distribute_kernel<<<dim3(2 * 64 * 64), dim3(256), 0, stream>>>(x, out);
}